// BlockCSR_56667798504236
// MI455X (gfx1250) — compile-verified
//
#include <hip/hip_runtime.h>
#include <hip/hip_bf16.h>
#include <math.h>

#define HIDDEN      2048
#define NUM_EXPERTS 8
#define TOP_K       2
#define BLOCKING    128
#define NBF         32            // FFN / BLOCKING = 4096 / 128

typedef __attribute__((ext_vector_type(2))) float v2f;
typedef __attribute__((ext_vector_type(8))) float v8f;

// ---------------------------------------------------------------------------
// Kernel 1: logits[N,8] = x[N,2048] @ W_gate.T  via V_WMMA_F32_16X16X4_F32.
// One wave per 16-token tile, 8 waves (256 threads) per block.
// W_gate (8x2048 fp32 = 64 KB) staged in LDS once per workgroup.
// Experts 8..15 of the 16-wide N tile are zeroed via a per-lane scale so the
// B fragment load stays unconditional (EXEC all-1s required by WMMA).
// ---------------------------------------------------------------------------
__global__ __launch_bounds__(256) void moe_gate_gemm(
    const float* __restrict__ x, const float* __restrict__ Wg,
    float* __restrict__ logits, int n_tokens)
{
  __shared__ float lw[NUM_EXPERTS * HIDDEN];  // 64 KB
  for (int i = threadIdx.x; i < NUM_EXPERTS * HIDDEN; i += 256)
    lw[i] = Wg[i];
  __syncthreads();

  const int wave = threadIdx.x >> 5;
  const int lane = threadIdx.x & 31;
  const int row0 = (blockIdx.x * 8 + wave) * 16;   // 16 tokens per wave
  if (row0 >= n_tokens) return;                    // wave-uniform

  // A fragment (16x4 f32): lane m = row, VGPR pair = K {k+2h, k+2h+1}, h=lane>=16
  const int m     = lane & 15;
  const int khalf = (lane >> 4) << 1;              // 0 or 2
  const float* xrow  = x + (size_t)(row0 + m) * HIDDEN + khalf;
  // B fragment (4x16 f32): lane n = expert column; columns >= 8 are zero
  const float* lwrow = lw + (m & (NUM_EXPERTS - 1)) * HIDDEN + khalf;
  const float  bscale = (m < NUM_EXPERTS) ? 1.0f : 0.0f;

  v8f c = {0.f, 0.f, 0.f, 0.f, 0.f, 0.f, 0.f, 0.f};

#pragma unroll 8
  for (int k = 0; k < HIDDEN; k += 4) {
    v2f a = *(const v2f*)(xrow + k);               // global_load_b64
    v2f b = *(const v2f*)(lwrow + k);              // ds_load_b64
    b.x *= bscale;
    b.y *= bscale;
    c = __builtin_amdgcn_wmma_f32_16x16x4_f32(
        /*neg_a=*/false, a, /*neg_b=*/false, b,
        /*c_mod=*/(short)0, c, /*reuse_a=*/false, /*reuse_b=*/false);
  }

  // D layout: VGPR j -> row M = j + 8*(lane>=16); column N = lane & 15
  const int nIdx = lane & 15;
  if (nIdx < NUM_EXPERTS) {
    const int mBase = row0 + ((lane >> 4) << 3);
#pragma unroll
    for (int j = 0; j < 8; ++j)
      logits[(size_t)(mBase + j) * NUM_EXPERTS + nIdx] = c[j];
  }
}

// ---------------------------------------------------------------------------
// Kernel 2: softmax over 8 logits, top-2 (ties -> lower index), renormalize.
// Softmax denominator cancels in the final renorm, so use raw exps.
// ---------------------------------------------------------------------------
__global__ __launch_bounds__(256) void moe_softmax_top2(
    const float* __restrict__ logits, float* __restrict__ weights,
    int* __restrict__ sel, int n_tokens)
{
  int t = blockIdx.x * 256 + threadIdx.x;
  if (t >= n_tokens) return;
  const float* lr = logits + (size_t)t * NUM_EXPERTS;
  float l[NUM_EXPERTS];
  float mx = -INFINITY;
#pragma unroll
  for (int e = 0; e < NUM_EXPERTS; ++e) { l[e] = lr[e]; mx = fmaxf(mx, l[e]); }
#pragma unroll
  for (int e = 0; e < NUM_EXPERTS; ++e) l[e] = __expf(l[e] - mx);

  int e0 = 0; float p0 = l[0];
#pragma unroll
  for (int e = 1; e < NUM_EXPERTS; ++e)
    if (l[e] > p0) { p0 = l[e]; e0 = e; }
  int e1 = (e0 == 0) ? 1 : 0; float p1 = l[e1];
#pragma unroll
  for (int e = 0; e < NUM_EXPERTS; ++e)
    if (e != e0 && l[e] > p1) { p1 = l[e]; e1 = e; }

  float inv = 1.0f / (p0 + p1);
  weights[2 * t]     = p0 * inv;
  weights[2 * t + 1] = p1 * inv;
  sel[2 * t]     = e0;
  sel[2 * t + 1] = e1;
}

// ---------------------------------------------------------------------------
// Kernel 3a: per-block expert histograms (256 assignments per block).
// ---------------------------------------------------------------------------
__global__ __launch_bounds__(256) void moe_hist(
    const int* __restrict__ sel, int* __restrict__ blockHist, int T)
{
  __shared__ int h[NUM_EXPERTS];
  if (threadIdx.x < NUM_EXPERTS) h[threadIdx.x] = 0;
  __syncthreads();
  int i = blockIdx.x * 256 + threadIdx.x;
  if (i < T) atomicAdd(&h[sel[i]], 1);
  __syncthreads();
  if (threadIdx.x < NUM_EXPERTS)
    blockHist[blockIdx.x * NUM_EXPERTS + threadIdx.x] = h[threadIdx.x];
}

// ---------------------------------------------------------------------------
// Kernel 3b: single-block scan. Produces per-(block,expert) scatter bases and
// the small metadata outputs: bins, block_bins, height_offsets, tokens/expert.
// ---------------------------------------------------------------------------
__global__ void moe_scan(
    const int* __restrict__ blockHist, int nBlocks,
    int* __restrict__ blockBase, int* __restrict__ bb_ws,
    int* __restrict__ out_bins, int* __restrict__ out_block_bins,
    int* __restrict__ out_height, int* __restrict__ out_tpe)
{
  __shared__ int cnt[NUM_EXPERTS], start[NUM_EXPERTS];
  int lane = threadIdx.x;
  if (lane < NUM_EXPERTS) {
    int run = 0;
    for (int b = 0; b < nBlocks; ++b) {
      blockBase[b * NUM_EXPERTS + lane] = run;
      run += blockHist[b * NUM_EXPERTS + lane];
    }
    cnt[lane] = run;
  }
  __syncthreads();
  if (lane == 0) {
    int acc = 0, bacc = 0, hacc = 0;
    for (int e = 0; e < NUM_EXPERTS; ++e) {
      start[e] = acc;
      int t = cnt[e];
      acc += t;
      int br = (t + BLOCKING - 1) / BLOCKING;
      bacc += br;
      hacc += br * BLOCKING;
      out_bins[e]       = acc;
      out_block_bins[e] = bacc;
      bb_ws[e]          = bacc;
      out_height[e]     = hacc;
      out_tpe[e]        = t;
    }
  }
  __syncthreads();
  if (lane < NUM_EXPERTS) {
    int s = start[lane];
    for (int b = 0; b < nBlocks; ++b) blockBase[b * NUM_EXPERTS + lane] += s;
  }
}

// ---------------------------------------------------------------------------
// Kernel 3c: stable scatter. Intra-wave rank via ballot/popcount (wave32),
// cross-wave prefix via LDS, per-block base from moe_scan.
// ---------------------------------------------------------------------------
__global__ __launch_bounds__(256) void moe_scatter(
    const int* __restrict__ sel, const int* __restrict__ blockBase,
    int* __restrict__ out_indices, int* __restrict__ out_bin_ids, int T)
{
  __shared__ int waveCnt[8][NUM_EXPERTS];
  const int lane = threadIdx.x & 31;
  const int wave = threadIdx.x >> 5;
  const int i = blockIdx.x * 256 + threadIdx.x;
  const int e = (i < T) ? sel[i] : -1;

  unsigned mymask = 0u;
  int myPopc = 0;
#pragma unroll
  for (int ex = 0; ex < NUM_EXPERTS; ++ex) {
    unsigned mk = (unsigned)__ballot(e == ex);
    if (e == ex)    mymask = mk;
    if (lane == ex) myPopc = __popc(mk);
  }
  if (lane < NUM_EXPERTS) waveCnt[wave][lane] = myPopc;
  __syncthreads();

  if (i < T) {
    int rank = __popc(mymask & ((1u << lane) - 1u));
    int base = 0;
    for (int w = 0; w < wave; ++w) base += waveCnt[w][e];
    int pos = blockBase[blockIdx.x * NUM_EXPERTS + e] + base + rank;
    out_indices[pos] = i;
    out_bin_ids[pos] = e;
  }
}

// ---------------------------------------------------------------------------
// Kernel 4: block-CSR topology. expert_of_rb = searchsorted(block_bins, rb, 'right').
// ---------------------------------------------------------------------------
__global__ __launch_bounds__(256) void moe_topology(
    const int* __restrict__ bb_ws, int* __restrict__ out_ffn,
    int* __restrict__ out_seq, int total)
{
  __shared__ int bb[NUM_EXPERTS];
  if (threadIdx.x < NUM_EXPERTS) bb[threadIdx.x] = bb_ws[threadIdx.x];
  __syncthreads();
  int idx = blockIdx.x * 256 + threadIdx.x;
  if (idx >= total) return;
  int rb = idx / NBF, col = idx % NBF;
  int e = 0;
#pragma unroll
  for (int j = 0; j < NUM_EXPERTS; ++j) e += (bb[j] <= rb) ? 1 : 0;
  if (e > NUM_EXPERTS - 1) e = NUM_EXPERTS - 1;
  bool valid = rb < bb[NUM_EXPERTS - 1];
  out_ffn[idx] = valid ? (e * NBF + col) : -1;
  out_seq[idx] = valid ? rb : -1;
}

// ---------------------------------------------------------------------------
extern "C" void kernel_launch(void* const* d_in, const int* in_sizes, int n_in,
                              void* d_out, int out_size, void* d_ws, size_t ws_size,
                              hipStream_t stream) {
  (void)n_in; (void)out_size; (void)ws_size;
  const float* x  = (const float*)d_in[0];
  const float* Wg = (const float*)d_in[1];

  const int n_tokens = in_sizes[0] / HIDDEN;          // 16384
  const int T        = n_tokens * TOP_K;              // 32768
  const int MB       = T / BLOCKING + NUM_EXPERTS;    // 264
  const int nBlocks  = (T + 255) / 256;               // 128

  // workspace layout
  char* ws = (char*)d_ws;
  float* logits   = (float*)ws;  ws += (size_t)n_tokens * NUM_EXPERTS * sizeof(float);
  int* sel        = (int*)ws;    ws += (size_t)T * sizeof(int);
  int* blockHist  = (int*)ws;    ws += (size_t)nBlocks * NUM_EXPERTS * sizeof(int);
  int* blockBase  = (int*)ws;    ws += (size_t)nBlocks * NUM_EXPERTS * sizeof(int);
  int* bb_ws      = (int*)ws;    ws += NUM_EXPERTS * sizeof(int);

  // output layout (flat, reference return order; int outputs keep int32 bits)
  float* o_weights    = (float*)d_out;
  int*   o_indices    = (int*)((float*)d_out + T);
  int*   o_bin_ids    = o_indices + T;
  int*   o_bins       = o_bin_ids + T;
  int*   o_block_bins = o_bins + NUM_EXPERTS;
  int*   o_height     = o_block_bins + NUM_EXPERTS;
  int*   o_tpe        = o_height + NUM_EXPERTS;
  int*   o_ffn        = o_tpe + NUM_EXPERTS;
  int*   o_seq        = o_ffn + MB * NBF;

  moe_gate_gemm<<<n_tokens / 128, 256, 0, stream>>>(x, Wg, logits, n_tokens);
  moe_softmax_top2<<<(n_tokens + 255) / 256, 256, 0, stream>>>(logits, o_weights, sel, n_tokens);
  moe_hist<<<nBlocks, 256, 0, stream>>>(sel, blockHist, T);
  moe_scan<<<1, 32, 0, stream>>>(blockHist, nBlocks, blockBase, bb_ws,
                                 o_bins, o_block_bins, o_height, o_tpe);
  moe_scatter<<<nBlocks, 256, 0, stream>>>(sel, blockBase, o_indices, o_bin_ids, T);
  moe_topology<<<(MB * NBF + 255) / 256, 256, 0, stream>>>(bb_ws, o_ffn, o_seq, MB * NBF);
}